// GATv3Operator_75874892251805
// MI455X (gfx1250) — compile-verified
//
#include <hip/hip_runtime.h>
#include <hip/hip_bf16.h>
#include <float.h>

// Problem constants (from reference): B=8, L=256, E=256, H=8, A=32
#define BB 8
#define LL 256
#define EE 256
#define HH 8
#define AA 32
#define QKV_N (3 * EE)       // 768
#define ML (BB * LL)         // 2048 rows
#define NEG_BIG (-3.4028234663852886e38f)

typedef float v2f __attribute__((ext_vector_type(2)));
typedef float v8f __attribute__((ext_vector_type(8)));
typedef int   v4i __attribute__((ext_vector_type(4)));

// address-space qualified int4 for the async-to-LDS builtin
typedef __attribute__((address_space(1))) v4i gv4i;   // global (prints as __device__)
typedef __attribute__((address_space(3))) v4i lv4i;   // LDS

__device__ __forceinline__ v8f wmma_f32(v2f a, v2f b, v8f c) {
    // D = A(16x4) x B(4x16) + C(16x16), all f32 (CDNA5 V_WMMA_F32_16X16X4_F32)
    return __builtin_amdgcn_wmma_f32_16x16x4_f32(
        /*neg_a=*/false, a, /*neg_b=*/false, b,
        /*c_mod=*/(short)0, c, /*reuse_a=*/false, /*reuse_b=*/false);
}

// ---- CDNA5 async global->LDS staging (ASYNCcnt path), guarded fallback ----
#if __has_builtin(__builtin_amdgcn_global_load_async_to_lds_b128)
#define HAS_ASYNC_LDS 1
#else
#define HAS_ASYNC_LDS 0
#endif

__device__ __forceinline__ void async_copy16(void* lds_dst, const void* gsrc) {
#if HAS_ASYNC_LDS
    __builtin_amdgcn_global_load_async_to_lds_b128(
        (gv4i*)(void*)gsrc, (lv4i*)lds_dst, /*offset=*/0, /*cpol=*/0);
#else
    *(float4*)lds_dst = *(const float4*)gsrc;
#endif
}

__device__ __forceinline__ void wait_async0() {
#if HAS_ASYNC_LDS
#if __has_builtin(__builtin_amdgcn_s_wait_asynccnt)
    __builtin_amdgcn_s_wait_asynccnt(0);
#else
    asm volatile("s_wait_asynccnt 0x0" ::: "memory");
#endif
#endif
}

// ---------------------------------------------------------------------------
// Kernel 1: qkv = X @ Wqkv^T   X:(2048,256) f32, Wqkv:(768,256) -> Y:(2048,768)
// Each wave computes a 16x64 strip: one A fragment feeds 4 B fragments.
// 128 M-tiles * 12 N-strips = 1536 wave-jobs -> 192 blocks of 8 waves.
// ---------------------------------------------------------------------------
__global__ __launch_bounds__(256) void qkv_gemm_kernel(
    const float* __restrict__ X, const float* __restrict__ W,
    float* __restrict__ Y) {
    const int lane = threadIdx.x & 31;
    const int wv   = threadIdx.x >> 5;
    const int job  = blockIdx.x * 8 + wv;          // [0, 1536)
    const int m0   = (job / (QKV_N / 64)) * 16;
    const int n0   = (job % (QKV_N / 64)) * 64;
    const int l15  = lane & 15;
    const int hi   = lane >> 4;

    // A fragment: lane l holds row M=l&15, K = k0 + 2*(l>>4) + {0,1}
    // B fragment: lane l holds col N=l&15, K = k0 + 2*(l>>4) + {0,1}  (B = W^T)
    const float* arow = X + (size_t)(m0 + l15) * EE + 2 * hi;
    const float* brow = W + (size_t)(n0 + l15) * EE + 2 * hi;

    v8f c0 = {}, c1 = {}, c2 = {}, c3 = {};
#pragma unroll 4
    for (int k0 = 0; k0 < EE; k0 += 4) {
        v2f a  = *(const v2f*)(arow + k0);
        v2f b0 = *(const v2f*)(brow + k0);
        v2f b1 = *(const v2f*)(brow + 16 * EE + k0);
        v2f b2 = *(const v2f*)(brow + 32 * EE + k0);
        v2f b3 = *(const v2f*)(brow + 48 * EE + k0);
        c0 = wmma_f32(a, b0, c0);
        c1 = wmma_f32(a, b1, c1);
        c2 = wmma_f32(a, b2, c2);
        c3 = wmma_f32(a, b3, c3);
    }
    // D: VGPR r -> row m0 + r + 8*hi, col n0 + nt*16 + (lane&15)
    float* yrow = Y + (size_t)(m0 + 8 * hi) * QKV_N + n0 + l15;
#pragma unroll
    for (int r = 0; r < 8; ++r) {
        yrow[(size_t)r * QKV_N +  0] = c0[r];
        yrow[(size_t)r * QKV_N + 16] = c1[r];
        yrow[(size_t)r * QKV_N + 32] = c2[r];
        yrow[(size_t)r * QKV_N + 48] = c3[r];
    }
}

// ---------------------------------------------------------------------------
// Kernel 2: per (b,h), 16-row chunk of i. 1024 blocks x 256 threads (8 waves).
//   scores[i][j] = sum_a Wa[a]*leaky(q[j,a]+k[i,a]); mask; softmax over j;
//   vals[i,:] = P[i,:] @ V  (f32 WMMA out of LDS, K split across wave pairs)
// ---------------------------------------------------------------------------
__global__ __launch_bounds__(256) void attn_kernel(
    const float* __restrict__ QKV,          // (2048, 768)
    const unsigned char* __restrict__ mask, // (B,1,L,L) bool
    const float* __restrict__ Wa,           // (32,)
    float* __restrict__ vals) {             // (B,L,E) with e = h*32 + a
    __shared__ float s_k[16][AA];           //  2 KB
    __shared__ float s_v[LL][AA];           // 32 KB
    __shared__ float s_p[16][LL];           // 16 KB
    __shared__ float s_part[4][16][AA];     //  8 KB  (58 KB total)

    const int tid  = threadIdx.x;
    const int lane = tid & 31;
    const int wv   = tid >> 5;
    const int bh   = blockIdx.x >> 4;   // 16 i-chunks per (b,h)
    const int iblk = blockIdx.x & 15;
    const int b    = bh >> 3;           // H = 8
    const int h    = bh & 7;
    const int i0   = iblk * 16;

    // stage K rows (this chunk) and full V (this head) via async global->LDS
    for (int t = tid; t < 16 * (AA / 4); t += 256) {       // 128 x b128
        int i = t >> 3, a4 = (t & 7) * 4;
        async_copy16(&s_k[i][a4],
                     QKV + (size_t)(b * LL + i0 + i) * QKV_N + h * 96 + 32 + a4);
    }
    for (int t = tid; t < LL * (AA / 4); t += 256) {       // 2048 x b128
        int j = t >> 3, a4 = (t & 7) * 4;
        async_copy16(&s_v[j][a4],
                     QKV + (size_t)(b * LL + j) * QKV_N + h * 96 + 64 + a4);
    }

    // each thread owns one column j = wv*32 + lane (8 waves * 32 lanes = 256)
    const int j = wv * 32 + lane;
    float qreg[AA], wreg[AA];
    {
        const float4* qp = (const float4*)(QKV + (size_t)(b * LL + j) * QKV_N + h * 96);
        const float4* wp = (const float4*)Wa;
#pragma unroll
        for (int t = 0; t < 8; ++t) {
            float4 q4 = qp[t], w4 = wp[t];
            qreg[4 * t + 0] = q4.x; qreg[4 * t + 1] = q4.y;
            qreg[4 * t + 2] = q4.z; qreg[4 * t + 3] = q4.w;
            wreg[4 * t + 0] = w4.x; wreg[4 * t + 1] = w4.y;
            wreg[4 * t + 2] = w4.z; wreg[4 * t + 3] = w4.w;
        }
    }
    wait_async0();
    __syncthreads();

    // masked additive-attention scores
    const size_t mrow = ((size_t)b * LL + i0) * LL + j;
    for (int i = 0; i < 16; ++i) {
        float acc = 0.f;
#pragma unroll
        for (int a = 0; a < AA; ++a) {
            float x = qreg[a] + s_k[i][a];
            x = (x >= 0.f) ? x : 0.1f * x;      // LeakyReLU(0.1)
            acc = fmaf(wreg[a], x, acc);
        }
        bool m = mask[mrow + (size_t)i * LL] != 0;
        s_p[i][j] = m ? NEG_BIG : acc;
    }
    __syncthreads();

    // softmax over j: wave wv handles rows 2*wv, 2*wv+1
    for (int rr = 0; rr < 2; ++rr) {
        const int i = wv * 2 + rr;
        float v[8];
        float mx = NEG_BIG;
#pragma unroll
        for (int u = 0; u < 8; ++u) { v[u] = s_p[i][lane + 32 * u]; mx = fmaxf(mx, v[u]); }
#pragma unroll
        for (int s = 16; s > 0; s >>= 1) mx = fmaxf(mx, __shfl_xor(mx, s, 32));
        float sum = 0.f;
#pragma unroll
        for (int u = 0; u < 8; ++u) { v[u] = __expf(v[u] - mx); sum += v[u]; }
#pragma unroll
        for (int s = 16; s > 0; s >>= 1) sum += __shfl_xor(sum, s, 32);
        const float inv = 1.0f / sum;
#pragma unroll
        for (int u = 0; u < 8; ++u) s_p[i][lane + 32 * u] = v[u] * inv;
    }
    __syncthreads();

    // P(16x256) @ V(256x32): wave w -> n-half (w&1), K-chunk 64 (w>>1).
    // All 8 waves run WMMA; partials reduced through LDS.
    {
        const int nh  = wv & 1;
        const int kc  = wv >> 1;            // 0..3
        const int l15 = lane & 15;
        const int hi  = lane >> 4;
        v8f c = {};
#pragma unroll 4
        for (int k0 = kc * 64; k0 < kc * 64 + 64; k0 += 4) {
            v2f a = *(const v2f*)(&s_p[l15][k0 + 2 * hi]);
            v2f bfrag;
            bfrag.x = s_v[k0 + 2 * hi + 0][nh * 16 + l15];
            bfrag.y = s_v[k0 + 2 * hi + 1][nh * 16 + l15];
            c = wmma_f32(a, bfrag, c);
        }
#pragma unroll
        for (int r = 0; r < 8; ++r) s_part[kc][r + 8 * hi][nh * 16 + l15] = c[r];
    }
    __syncthreads();

    // reduce the 4 K-chunk partials and store vals (B,L,E)
    for (int t = tid; t < 16 * AA; t += 256) {
        int i = t >> 5, a = t & 31;
        float s = s_part[0][i][a] + s_part[1][i][a] + s_part[2][i][a] + s_part[3][i][a];
        vals[(size_t)(b * LL + i0 + i) * EE + h * AA + a] = s;
    }
}

// ---------------------------------------------------------------------------
// Kernel 3: out = vals @ Wout^T   vals:(2048,256), Wout:(256,256) -> (2048,256)
// 16x64 strip per wave: 128 M-tiles * 4 N-strips = 512 jobs -> 64 blocks.
// ---------------------------------------------------------------------------
__global__ __launch_bounds__(256) void out_gemm_kernel(
    const float* __restrict__ Xv, const float* __restrict__ W,
    float* __restrict__ Y) {
    const int lane = threadIdx.x & 31;
    const int wv   = threadIdx.x >> 5;
    const int job  = blockIdx.x * 8 + wv;          // [0, 512)
    const int m0   = (job / (EE / 64)) * 16;
    const int n0   = (job % (EE / 64)) * 64;
    const int l15  = lane & 15;
    const int hi   = lane >> 4;

    const float* arow = Xv + (size_t)(m0 + l15) * EE + 2 * hi;
    const float* brow = W  + (size_t)(n0 + l15) * EE + 2 * hi;

    v8f c0 = {}, c1 = {}, c2 = {}, c3 = {};
#pragma unroll 4
    for (int k0 = 0; k0 < EE; k0 += 4) {
        v2f a  = *(const v2f*)(arow + k0);
        v2f b0 = *(const v2f*)(brow + k0);
        v2f b1 = *(const v2f*)(brow + 16 * EE + k0);
        v2f b2 = *(const v2f*)(brow + 32 * EE + k0);
        v2f b3 = *(const v2f*)(brow + 48 * EE + k0);
        c0 = wmma_f32(a, b0, c0);
        c1 = wmma_f32(a, b1, c1);
        c2 = wmma_f32(a, b2, c2);
        c3 = wmma_f32(a, b3, c3);
    }
    float* yrow = Y + (size_t)(m0 + 8 * hi) * EE + n0 + l15;
#pragma unroll
    for (int r = 0; r < 8; ++r) {
        yrow[(size_t)r * EE +  0] = c0[r];
        yrow[(size_t)r * EE + 16] = c1[r];
        yrow[(size_t)r * EE + 32] = c2[r];
        yrow[(size_t)r * EE + 48] = c3[r];
    }
}

// ---------------------------------------------------------------------------
extern "C" void kernel_launch(void* const* d_in, const int* in_sizes, int n_in,
                              void* d_out, int out_size, void* d_ws, size_t ws_size,
                              hipStream_t stream) {
    const float*         emb   = (const float*)d_in[0];          // (B,L,E)
    const unsigned char* amask = (const unsigned char*)d_in[1];  // (B,1,L,L) bool
    const float*         w_qkv = (const float*)d_in[2];          // (768,256)
    const float*         w_att = (const float*)d_in[3];          // (32,)
    const float*         w_out = (const float*)d_in[4];          // (256,256)
    float*               out   = (float*)d_out;                  // (B,L,E)

    float* qkv_buf  = (float*)d_ws;                    // 2048*768 f32 (6 MB)
    float* vals_buf = qkv_buf + (size_t)ML * QKV_N;    // 2048*256 f32 (2 MB)

    // 1) QKV projection: 1536 wave-jobs / 8 = 192 blocks
    qkv_gemm_kernel<<<192, 256, 0, stream>>>(emb, w_qkv, qkv_buf);
    // 2) fused additive attention + softmax + P@V: B*H*(L/16) = 1024 blocks
    attn_kernel<<<BB * HH * (LL / 16), 256, 0, stream>>>(qkv_buf, amask, w_att, vals_buf);
    // 3) output projection: 512 wave-jobs / 8 = 64 blocks
    out_gemm_kernel<<<64, 256, 0, stream>>>(vals_buf, w_out, out);
}